// CoPE_44641890074920
// MI455X (gfx1250) — compile-verified
//
#include <hip/hip_runtime.h>

// CoPE fused kernel for MI455X (gfx1250, wave32).
// B=1, H=16, S=2048, D=64, NPOS=64.

typedef __attribute__((ext_vector_type(2))) float v2f;
typedef __attribute__((ext_vector_type(8))) float v8f;

#define CP_H    16
#define CP_S    2048
#define CP_D    64
#define CP_NPOS 64

__device__ __forceinline__ float cope_sigmoid(float x) {
    return 1.0f / (1.0f + __expf(-x));
}

// pos >= 0 always (sum of sigmoids). Gather+interp against 64-entry LDS row.
__device__ __forceinline__ float cope_interp(float p, const float* __restrict__ prow) {
    p = fminf(p, 63.0f);
    float f = floorf(p);
    int i0 = (int)f;
    int i1 = i0 + 1;
    if (i1 > 63) i1 = 63;
    float w  = p - f;                 // fractional weight
    float lf = prow[i0];              // logits_floor
    float lc = prow[i1];              // logits_ceil (w==0 when it would differ)
    return lc * w + lf * (1.0f - w);
}

__launch_bounds__(128, 1)
__global__ void cope_fused_kernel(const float* __restrict__ q,     // [H,S,D]
                                  const float* __restrict__ attn,  // [H,S,S]
                                  const float* __restrict__ pe,    // [NPOS,H*D]
                                  float* __restrict__ out) {       // [H,S,S]
    __shared__ float sQ[16][CP_D];          // 4 KB  query tile
    __shared__ float sPE[CP_NPOS][CP_D];    // 16 KB pe slice for this head
    __shared__ float sP[16][CP_NPOS];       // 4 KB  logits_int tile

    const int tid  = threadIdx.x;
    const int lane = tid & 31;
    const int wave = tid >> 5;
    const int bid  = blockIdx.x;
    const int h    = bid >> 7;              // / (S/16)
    const int s0   = (bid & 127) << 4;      // tile start row

    // ---------------- Phase 0: stage Q tile and PE slice into LDS ----------
    {
        const float4* qbase = (const float4*)(q + (size_t)(h * CP_S + s0) * CP_D);
        #pragma unroll
        for (int i = tid; i < 16 * CP_D / 4; i += 128)
            ((float4*)&sQ[0][0])[i] = qbase[i];

        // pe element: pe[n*(H*D) + h*D + d]
        #pragma unroll
        for (int i = tid; i < CP_NPOS * CP_D / 4; i += 128) {
            int n  = i >> 4;                // 16 float4 per 64-float row
            int d4 = i & 15;
            ((float4*)&sPE[0][0])[i] =
                ((const float4*)(pe + (size_t)n * (CP_H * CP_D) + h * CP_D))[d4];
        }
    }
    __syncthreads();

    // ---------------- Phase 1: logits_int tile via f32 WMMA ----------------
    // Each wave computes one 16x16 N-tile: C = Qtile(16x64) * PE^T(64x16)
    // using 16 chained V_WMMA_F32_16X16X4_F32 (K=4 each).
    {
        const int n0    = wave << 4;
        const int m     = lane & 15;
        const int khalf = (lane >> 4) << 1;     // 0 or 2: K pair this half holds
        v8f c = {};
        #pragma unroll
        for (int k0 = 0; k0 < CP_D; k0 += 4) {
            v2f a, b;
            a.x = sQ[m][k0 + khalf];
            a.y = sQ[m][k0 + khalf + 1];
            // B[k][n] = pe_h[n0+n][k]
            b.x = sPE[n0 + m][k0 + khalf];
            b.y = sPE[n0 + m][k0 + khalf + 1];
            c = __builtin_amdgcn_wmma_f32_16x16x4_f32(false, a, false, b,
                                                      (short)0, c, false, false);
        }
        // C layout: VGPR r, lanes 0-15 -> row r, lanes 16-31 -> row r+8; col = lane&15
        #pragma unroll
        for (int r = 0; r < 8; ++r) {
            int row = r + ((lane >> 4) << 3);
            sP[row][n0 + (lane & 15)] = c[r];
        }
    }
    __syncthreads();

    // ---------------- Phase 2: streaming reverse-cumsum + gather-interp ----
    // Each wave owns 4 rows; per row walk 2048 keys backwards in 128-wide
    // chunks (float4 per lane). Single pass over attn_logits.
    for (int rr = 0; rr < 4; ++rr) {
        const int row = (wave << 2) + rr;
        const int s   = s0 + row;
        const float* __restrict__ arow = attn + (size_t)(h * CP_S + s) * CP_S;
        float* __restrict__       orow = out  + (size_t)(h * CP_S + s) * CP_S;
        const float* __restrict__ prow = &sP[row][0];

        float carry = 0.0f;
        for (int chunk = (CP_S / 128) - 1; chunk >= 0; --chunk) {
            const int base = chunk * 128 + lane * 4;
            if (chunk > 0)
                __builtin_prefetch(arow + base - 128, 0, 0);   // global_prefetch_b8

            float4 g = *(const float4*)(arow + base);
            float g0 = cope_sigmoid(g.x);
            float g1 = cope_sigmoid(g.y);
            float g2 = cope_sigmoid(g.z);
            float g3 = cope_sigmoid(g.w);
            float lsum = g0 + g1 + g2 + g3;

            // wave32 inclusive suffix scan: suf[t] = sum_{t'>=t} lsum[t']
            float suf = lsum;
            #pragma unroll
            for (int off = 1; off < 32; off <<= 1) {
                float o = __shfl_down(suf, off, 32);
                suf += (lane + off < 32) ? o : 0.0f;
            }
            float total = __shfl(suf, 0, 32);   // full chunk sum
            float exc   = suf - lsum;           // suffix of lanes strictly above

            // element j (global idx base+j): pos = carry + exc + sum_{j'>=j} g[j']
            float p3 = carry + exc + g3;
            float p2 = p3 + g2;
            float p1 = p2 + g1;
            float p0 = p1 + g0;

            float4 o4;
            o4.x = cope_interp(p0, prow);
            o4.y = cope_interp(p1, prow);
            o4.z = cope_interp(p2, prow);
            o4.w = cope_interp(p3, prow);
            *(float4*)(orow + base) = o4;

            carry += total;
        }
    }
}

extern "C" void kernel_launch(void* const* d_in, const int* in_sizes, int n_in,
                              void* d_out, int out_size, void* d_ws, size_t ws_size,
                              hipStream_t stream) {
    (void)in_sizes; (void)n_in; (void)d_ws; (void)ws_size; (void)out_size;
    const float* q    = (const float*)d_in[0];   // [1,16,2048,64]
    const float* attn = (const float*)d_in[1];   // [1,16,2048,2048]
    const float* pe   = (const float*)d_in[2];   // [1,64,1024]
    float* out = (float*)d_out;                  // [1,16,2048,2048]

    dim3 grid(CP_H * (CP_S / 16));               // 2048 workgroups
    dim3 block(128);                             // 4 wave32 waves
    cope_fused_kernel<<<grid, block, 0, stream>>>(q, attn, pe, out);
}